// Encoder_Postnet_5506148073942
// MI455X (gfx1250) — compile-verified
//
#include <hip/hip_runtime.h>

#define NB       32
#define T        4096
#define PHONE    600
#define HDIM     256

typedef __attribute__((ext_vector_type(2))) float v2f;
typedef __attribute__((ext_vector_type(8))) float v8f;

// ---------------------------------------------------------------------------
// Kernel 1: per-batch prefix sum of phone-change flags, clamped to PHONE-1.
// One block per batch; 256 threads x 16 elements = 4096.
// ---------------------------------------------------------------------------
__global__ __launch_bounds__(256) void idx_scan_kernel(const int* __restrict__ align,
                                                       int* __restrict__ idxbuf) {
    __shared__ int partial[256];
    const int b   = blockIdx.x;
    const int tid = threadIdx.x;
    const int* a  = align + b * T;
    const int base = tid * 16;

    int prev = (base == 0) ? 0 : a[base - 1];
    int run[16];
    int s = 0;
#pragma unroll
    for (int i = 0; i < 16; ++i) {
        int cur = a[base + i];
        s += (cur != prev) ? 1 : 0;
        run[i] = s;
        prev = cur;
    }
    partial[tid] = s;
    __syncthreads();

    // Hillis-Steele inclusive scan over the 256 per-thread sums.
#pragma unroll
    for (int off = 1; off < 256; off <<= 1) {
        int v = (tid >= off) ? partial[tid - off] : 0;
        __syncthreads();
        partial[tid] += v;
        __syncthreads();
    }
    const int excl = partial[tid] - s;

    int* dst = idxbuf + b * T + base;
#pragma unroll
    for (int i = 0; i < 16; ++i) {
        int v = excl + run[i];
        dst[i] = (v > (PHONE - 1)) ? (PHONE - 1) : v;   // JAX clamps OOB gather
    }
}

// ---------------------------------------------------------------------------
// Kernel 2: fused gather + rank-4 update via V_WMMA_F32_16X16X4_F32.
//   D(16x16) = A(16x4) x B(4x16) + C(16x16)
//   A rows  = t in tile, cols = [pitch, beats, t/T, 1]
//   B cols  = h in tile, rows = [w_pitch, w_beats, w_pos, b_pitch+b_beats+b_pos]
//   C       = gathered encoder rows enc[b, idx[t], h-tile]
// 8 waves / block; each wave owns one 16-t tile and loops 16 h-tiles (H=256).
// Base pointers are pre-biased by the lane index so every per-tile access is
// base + compile-time immediate offset (no per-tile 64-bit address math).
// ---------------------------------------------------------------------------
__global__ __launch_bounds__(256) void fuse_wmma_kernel(
    const float* __restrict__ enc,
    const float* __restrict__ pitch,
    const float* __restrict__ beats,
    const float* __restrict__ w_pitch,
    const float* __restrict__ b_pitch,
    const float* __restrict__ w_beats,
    const float* __restrict__ b_beats,
    const float* __restrict__ w_pos,
    const float* __restrict__ b_pos,
    const int*   __restrict__ idxbuf,
    float*       __restrict__ out) {

    const int lane = threadIdx.x & 31;             // wave32
    const int wave = threadIdx.x >> 5;
    const int tilesPerBatch = T / 128;             // 32
    const int b  = blockIdx.x / tilesPerBatch;
    const int tb = blockIdx.x % tilesPerBatch;
    const int t0 = tb * 128 + wave * 16;
    const bool hi = (lane >= 16);
    const int  l  = lane & 15;
    const int  tA = t0 + l;                        // A-matrix row M = l for both halves

    // A operand: lanes 0-15 carry K0,K1 = (pitch, beats); lanes 16-31 carry K2,K3 = (t/T, 1)
    v2f A;
    if (!hi) {
        A.x = __builtin_nontemporal_load(&pitch[b * T + tA]);
        A.y = __builtin_nontemporal_load(&beats[b * T + tA]);
    } else {
        A.x = (float)tA * (1.0f / (float)T);
        A.y = 1.0f;
    }

    // Preload all 16 B operands (loop-invariant weights) once.
    v2f Ball[HDIM / 16];
    if (!hi) {
#pragma unroll
        for (int ht = 0; ht < HDIM / 16; ++ht) {
            const int h = ht * 16 + l;
            Ball[ht].x = w_pitch[h];
            Ball[ht].y = w_beats[h];
        }
    } else {
#pragma unroll
        for (int ht = 0; ht < HDIM / 16; ++ht) {
            const int h = ht * 16 + l;
            Ball[ht].x = w_pos[h];
            Ball[ht].y = b_pitch[h] + b_beats[h] + b_pos[h];
        }
    }

    // Resolve the 8 gathered encoder row bases for this half-wave's C rows,
    // pre-biased by l so inner-loop offsets are pure immediates (ht*64 bytes).
    const int* idxrow = idxbuf + b * T + t0 + (hi ? 8 : 0);
    const float* rowp[8];
#pragma unroll
    for (int r = 0; r < 8; ++r) {
        const int ridx = __builtin_nontemporal_load(&idxrow[r]);  // clamped in kernel 1
        rowp[r] = enc + ((size_t)b * PHONE + (size_t)ridx) * HDIM + l;
    }

    // Output base, also pre-biased by l: store offsets are r*1024 + ht*64 bytes.
    float* outrow = out + ((size_t)b * T + (size_t)(t0 + (hi ? 8 : 0))) * HDIM + l;

#pragma unroll
    for (int ht = 0; ht < HDIM / 16; ++ht) {
        // C operand: gathered encoder tile (regular-temporal: high L2/WGP$ reuse).
        v8f C;
#pragma unroll
        for (int r = 0; r < 8; ++r) C[r] = rowp[r][ht * 16];

        // D = A x B + C   (v_wmma_f32_16x16x4_f32; EXEC is all-1s here)
        v8f D = __builtin_amdgcn_wmma_f32_16x16x4_f32(
            /*neg_a=*/false, A, /*neg_b=*/false, Ball[ht],
            /*c_mod=*/(short)0, C, /*reuse_a=*/false, /*reuse_b=*/false);

        // Output is a 128 MiB write-once stream: non-temporal so it does not
        // evict the reused encoder table from the 192 MB L2.
#pragma unroll
        for (int r = 0; r < 8; ++r)
            __builtin_nontemporal_store(D[r], &outrow[r * HDIM + ht * 16]);
    }
}

// ---------------------------------------------------------------------------
// Host-side launcher.
// Input order: encoder_out, pitch, beats, w_pitch, b_pitch, w_beats, b_beats,
//              w_pos, b_pos, align_phone
// ---------------------------------------------------------------------------
extern "C" void kernel_launch(void* const* d_in, const int* in_sizes, int n_in,
                              void* d_out, int out_size, void* d_ws, size_t ws_size,
                              hipStream_t stream) {
    const float* enc     = (const float*)d_in[0];
    const float* pitch   = (const float*)d_in[1];
    const float* beats   = (const float*)d_in[2];
    const float* w_pitch = (const float*)d_in[3];
    const float* b_pitch = (const float*)d_in[4];
    const float* w_beats = (const float*)d_in[5];
    const float* b_beats = (const float*)d_in[6];
    const float* w_pos   = (const float*)d_in[7];
    const float* b_pos   = (const float*)d_in[8];
    const int*   align   = (const int*)d_in[9];
    float*       out     = (float*)d_out;
    int*         idxbuf  = (int*)d_ws;             // NB*T ints = 512 KB scratch

    idx_scan_kernel<<<dim3(NB), dim3(256), 0, stream>>>(align, idxbuf);

    fuse_wmma_kernel<<<dim3(NB * (T / 128)), dim3(256), 0, stream>>>(
        enc, pitch, beats, w_pitch, b_pitch, w_beats, b_beats, w_pos, b_pos,
        idxbuf, out);
}